// PREFFFT_12549894439512
// MI455X (gfx1250) — compile-verified
//
#include <hip/hip_runtime.h>

// PREF / factored-Fourier plane sampler for MI455X (gfx1250, wave32).
//
// Planes: 3 x (256ch x 256 x 256) f32 = 192 MB total == MI455X L2, so all
// gathers are L2-resident after warmup -> kernel is L2-request-rate bound.
// Per point per (plane, channel): 30 row-pairs (8 B each) gathered + tiny
// Fourier contraction. We minimize requests with paired-column 8B loads and
// (primary path) stage them through the CDNA5 async-to-LDS engine:
//   issue tile k+1 (30 global_load_async_to_lds_b64, per-lane addresses)
//   s_wait_asynccnt 30   (async loads complete in order -> tile k landed)
//   compute tile k from LDS (conflict-free ds_load_b64, lane stride 8 B)
// Double-buffered, channel-outer/plane-inner so the accumulator is scalar.

#define PREF_CH 16
#define PREF_RR 8
#define PREF_CHSTRIDE 65536            // 256*256 elements between channels
#define TILE_ELEMS (PREF_RR * PREF_CHSTRIDE)   // elements between channel-tiles

// ---------------- CDNA5 async-to-LDS availability probe -------------------
#if __has_builtin(__builtin_amdgcn_global_load_async_to_lds_b64) && \
    __has_builtin(__builtin_amdgcn_s_wait_asynccnt)
#define PREF_ASYNC 1
#else
#define PREF_ASYNC 0
#endif

typedef __attribute__((ext_vector_type(4))) float v4f;
// Builtin's operand type per clang diagnostic: pointer to <2 x i32>.
typedef int v2i __attribute__((ext_vector_type(2)));
typedef __attribute__((address_space(1))) v2i gv2i_t;   // global source
typedef __attribute__((address_space(3))) v2i lv2i_t;   // LDS destination

// 8-byte pair, 4-byte alignment (x0 may be odd)
struct __attribute__((aligned(4))) f2v { float a, b; };
// 8-byte-aligned LDS pair (slot bases are 8B aligned)
struct __attribute__((aligned(8))) d2v { float x, y; };

// Raw hardware trig: v_sin_f32/v_cos_f32 take input in revolutions
// (sin(2*pi*x)), valid |x| <= 256. Our max is 255*64/256 = 63.75.
__device__ __forceinline__ float cos2pi(float rev) { return __builtin_amdgcn_cosf(rev); }
__device__ __forceinline__ float sin2pi(float rev) { return __builtin_amdgcn_sinf(rev); }

struct Tap { int i; float wl, wr; };

// align_corners=True: ix = (g+1)*0.5*(256-1); border clamp folded into weights
__device__ __forceinline__ Tap prep(float g) {
    float ix = fmaf(g, 127.5f, 127.5f);
    float f  = floorf(ix);
    float w  = ix - f;
    int   i0 = (int)f;
    if (i0 < 0)   { i0 = 0;   w = 0.0f; }
    if (i0 > 254) { i0 = 254; w = 1.0f; }   // ix==255 -> use col 255 only
    Tap t; t.i = i0; t.wl = 1.0f - w; t.wr = w;
    return t;
}

__device__ __forceinline__ void make_basis(float cf, float* cb, float* sb) {
    cb[0] = 1.0f; sb[0] = 0.0f;            // coef[0] == 0
    float rev = cf * (1.0f / 256.0f);      // revolutions: cf * 2^(r-1)/256
    #pragma unroll
    for (int r = 1; r < PREF_RR; ++r) {
        cb[r] = cos2pi(rev);
        sb[r] = sin2pi(rev);
        rev = rev + rev;
    }
}

#if PREF_ASYNC
// =========================== async-to-LDS path ============================

#define TPB 128
#define NSLOT 30                          // 16 real + 14 imag row-pairs
#define SLOT_STRIDE (TPB * 8)             // bytes; lane-interleaved slots
#define BUF_BYTES (NSLOT * SLOT_STRIDE)   // 30720 B per buffer

__device__ __forceinline__ void async_b64(const float* g, char* l) {
    // src global (AS1 <2 x i32>*), dst LDS (AS3 <2 x i32>*), imm offset 0
    // (applies to BOTH addresses per ISA, so keep 0), cpol 0 (RT: planes
    // must stay L2-cached).
    __builtin_amdgcn_global_load_async_to_lds_b64((gv2i_t*)g, (lv2i_t*)l, 0, 0);
}

__device__ __forceinline__ void wait_ds0() {
#if __has_builtin(__builtin_amdgcn_s_wait_dscnt)
    __builtin_amdgcn_s_wait_dscnt(0);
#else
    asm volatile("s_wait_dscnt 0x0" ::: "memory");
#endif
    asm volatile("" ::: "memory");
}

// Issue one (plane, channel) tile: 30 x 8B gathers into this thread's slots.
__device__ __forceinline__ void issue_tile(const float* tile, char* lbase) {
    const float* pim = tile + 128 * PREF_CHSTRIDE;
    #pragma unroll
    for (int r = 0; r < PREF_RR; ++r) {
        async_b64(tile + r * PREF_CHSTRIDE,       lbase + (2 * r    ) * SLOT_STRIDE);
        async_b64(tile + r * PREF_CHSTRIDE + 256, lbase + (2 * r + 1) * SLOT_STRIDE);
    }
    #pragma unroll
    for (int r = 1; r < PREF_RR; ++r) {   // r=0: sin(0)=0, no imag loads
        async_b64(pim + r * PREF_CHSTRIDE,        lbase + (14 + 2 * r) * SLOT_STRIDE);
        async_b64(pim + r * PREF_CHSTRIDE + 256,  lbase + (15 + 2 * r) * SLOT_STRIDE);
    }
}

// Consume one tile from LDS: bilinear + Fourier dot -> one scalar.
__device__ __forceinline__ float compute_tile(const char* lbase,
        const float* cb, const float* sb,
        float wxl, float wxr, float wyl, float wyr) {
    float s = 0.0f;
    #pragma unroll
    for (int r = 0; r < PREF_RR; ++r) {
        d2v a = *(const d2v*)(lbase + (2 * r    ) * SLOT_STRIDE);
        d2v b = *(const d2v*)(lbase + (2 * r + 1) * SLOT_STRIDE);
        float v = fmaf(wyr, fmaf(wxr, b.y, wxl * b.x),
                       wyl * fmaf(wxr, a.y, wxl * a.x));
        s = fmaf(cb[r], v, s);
    }
    #pragma unroll
    for (int r = 1; r < PREF_RR; ++r) {
        d2v a = *(const d2v*)(lbase + (14 + 2 * r) * SLOT_STRIDE);
        d2v b = *(const d2v*)(lbase + (15 + 2 * r) * SLOT_STRIDE);
        float v = fmaf(wyr, fmaf(wxr, b.y, wxl * b.x),
                       wyl * fmaf(wxr, a.y, wxl * a.x));
        s = fmaf(-sb[r], v, s);
    }
    return s;
}

__global__ __launch_bounds__(TPB) void PREFFFT_12549894439512_kernel(
    const float* __restrict__ xyz,
    const float* __restrict__ Pu,
    const float* __restrict__ Pv,
    const float* __restrict__ Pw,
    float* __restrict__ out, int N)
{
    __shared__ char smem[2 * BUF_BYTES];

    int n = blockIdx.x * TPB + threadIdx.x;
    if (n >= N) return;                    // N % 128 == 0: whole waves

    float c0 = xyz[3 * n + 0];
    float c1 = xyz[3 * n + 1];
    float c2 = xyz[3 * n + 2];
    Tap t0 = prep(c0), t1 = prep(c1), t2 = prep(c2);

    // su: Pu(x<-c1, y<-c2) int c0 ; sv: Pv(x<-c0, y<-c2) int c1 ;
    // sw: Pw(x<-c0, y<-c1) int c2
    const float* bu = Pu + t2.i * 256 + t1.i;
    const float* bv = Pv + t2.i * 256 + t0.i;
    const float* bw = Pw + t1.i * 256 + t0.i;

    float cbu[PREF_RR], sbu[PREF_RR], cbv[PREF_RR], sbv[PREF_RR],
          cbw[PREF_RR], sbw[PREF_RR];
    make_basis(fmaf(c0, 127.5f, 127.5f), cbu, sbu);
    make_basis(fmaf(c1, 127.5f, 127.5f), cbv, sbv);
    make_basis(fmaf(c2, 127.5f, 127.5f), cbw, sbw);

    char* cur = smem + threadIdx.x * 8;               // buffer 0 slot base
    char* nxt = smem + BUF_BYTES + threadIdx.x * 8;   // buffer 1 slot base

    issue_tile(bu, cur);                              // prime: (u, ch0)

    #pragma unroll 1
    for (int ch = 0; ch < PREF_CH; ++ch) {
        // ---- plane u tile ----
        wait_ds0();
        issue_tile(bv + ch * TILE_ELEMS, nxt);
        __builtin_amdgcn_s_wait_asynccnt(NSLOT);      // u tile landed
        float su = compute_tile(cur, cbu, sbu, t1.wl, t1.wr, t2.wl, t2.wr);
        { char* t = cur; cur = nxt; nxt = t; }

        // ---- plane v tile ----
        wait_ds0();
        issue_tile(bw + ch * TILE_ELEMS, nxt);
        __builtin_amdgcn_s_wait_asynccnt(NSLOT);      // v tile landed
        float sv = compute_tile(cur, cbv, sbv, t0.wl, t0.wr, t2.wl, t2.wr);
        { char* t = cur; cur = nxt; nxt = t; }

        // ---- plane w tile ----
        wait_ds0();
        if (ch < PREF_CH - 1) {
            issue_tile(bu + (ch + 1) * TILE_ELEMS, nxt);
            __builtin_amdgcn_s_wait_asynccnt(NSLOT);  // w tile landed
        } else {
            __builtin_amdgcn_s_wait_asynccnt(0);      // drain
        }
        float sw = compute_tile(cur, cbw, sbw, t0.wl, t0.wr, t1.wl, t1.wr);
        { char* t = cur; cur = nxt; nxt = t; }

        // Nontemporal store: keep L2 reserved for the planes.
        __builtin_nontemporal_store(su + sv + sw, out + (size_t)n * PREF_CH + ch);
    }
}

#define PREF_TPB TPB

#else
// ======================= fallback: direct-load path =======================

__device__ __forceinline__ float bilin(const float* __restrict__ p,
                                       float wxl, float wxr, float wyl, float wyr) {
    f2v r0 = *(const f2v*)p;
    f2v r1 = *(const f2v*)(p + 256);
    float top = fmaf(wxr, r0.b, wxl * r0.a);
    float bot = fmaf(wxr, r1.b, wxl * r1.a);
    return fmaf(wyr, bot, wyl * top);
}

__device__ __forceinline__ void plane_accum(
    const float* __restrict__ base, float wxl, float wxr, float wyl, float wyr,
    float cf, float* __restrict__ acc)
{
    float cb[PREF_RR], sb[PREF_RR];
    make_basis(cf, cb, sb);
    #pragma unroll 2
    for (int ch = 0; ch < PREF_CH; ++ch) {
        const float* pre = base + ch * TILE_ELEMS;
        const float* pim = pre + 128 * PREF_CHSTRIDE;
        float sre = 0.0f;
        #pragma unroll
        for (int r = 0; r < PREF_RR; ++r)
            sre = fmaf(cb[r], bilin(pre + r * PREF_CHSTRIDE, wxl, wxr, wyl, wyr), sre);
        float sim = 0.0f;
        #pragma unroll
        for (int r = 1; r < PREF_RR; ++r)
            sim = fmaf(sb[r], bilin(pim + r * PREF_CHSTRIDE, wxl, wxr, wyl, wyr), sim);
        acc[ch] += sre - sim;
    }
}

__global__ __launch_bounds__(256) void PREFFFT_12549894439512_kernel(
    const float* __restrict__ xyz,
    const float* __restrict__ Pu,
    const float* __restrict__ Pv,
    const float* __restrict__ Pw,
    float* __restrict__ out, int N)
{
    int n = blockIdx.x * 256 + threadIdx.x;
    if (n >= N) return;
    float c0 = xyz[3 * n + 0];
    float c1 = xyz[3 * n + 1];
    float c2 = xyz[3 * n + 2];
    Tap t0 = prep(c0), t1 = prep(c1), t2 = prep(c2);

    float acc[PREF_CH];
    #pragma unroll
    for (int i = 0; i < PREF_CH; ++i) acc[i] = 0.0f;

    plane_accum(Pu + t2.i * 256 + t1.i, t1.wl, t1.wr, t2.wl, t2.wr,
                fmaf(c0, 127.5f, 127.5f), acc);
    plane_accum(Pv + t2.i * 256 + t0.i, t0.wl, t0.wr, t2.wl, t2.wr,
                fmaf(c1, 127.5f, 127.5f), acc);
    plane_accum(Pw + t1.i * 256 + t0.i, t0.wl, t0.wr, t1.wl, t1.wr,
                fmaf(c2, 127.5f, 127.5f), acc);

    v4f* o = (v4f*)(out + (size_t)n * PREF_CH);
    #pragma unroll
    for (int q = 0; q < 4; ++q) {
        v4f t = {acc[4 * q], acc[4 * q + 1], acc[4 * q + 2], acc[4 * q + 3]};
        __builtin_nontemporal_store(t, o + q);
    }
}

#define PREF_TPB 256

#endif

extern "C" void kernel_launch(void* const* d_in, const int* in_sizes, int n_in,
                              void* d_out, int out_size, void* d_ws, size_t ws_size,
                              hipStream_t stream) {
    const float* xyz = (const float*)d_in[0];
    const float* Pu  = (const float*)d_in[1];
    const float* Pv  = (const float*)d_in[2];
    const float* Pw  = (const float*)d_in[3];
    float* out = (float*)d_out;
    int N = in_sizes[0] / 3;   // (N,3) coords

    dim3 block(PREF_TPB);
    dim3 grid((N + PREF_TPB - 1) / PREF_TPB);
    hipLaunchKernelGGL(PREFFFT_12549894439512_kernel, grid, block, 0, stream,
                       xyz, Pu, Pv, Pw, out, N);
}